// GATNet_53197464928683
// MI455X (gfx1250) — compile-verified
//
#include <hip/hip_runtime.h>
#include <math.h>

typedef float v2f __attribute__((ext_vector_type(2)));
typedef float v8f __attribute__((ext_vector_type(8)));

namespace {
enum : int {
  B_ = 4,
  N_TOTAL_ = 40000,
  N_SRC0_ = 30000, N_DST0_ = 15000, E0_ = 360000,
  N_DST1_ = 7500,  E1_ = 180000,
  D_IN_ = 64,
  H1_ = 8, C1_ = 8,
  H2_ = 1, C2_ = 32,
};
constexpr float NEG_SLOPE_ = 0.2f;
}

// ---- order-preserving float<->uint encoding for atomic max ----
__device__ __forceinline__ unsigned fenc(float f) {
  unsigned b = __float_as_uint(f);
  return (b & 0x80000000u) ? ~b : (b | 0x80000000u);
}
__device__ __forceinline__ float fdec(unsigned k) {
  unsigned b = (k & 0x80000000u) ? (k & 0x7fffffffu) : ~k;
  return __uint_as_float(b);
}

// ============================================================
// WMMA fp32 GEMM with row gather:  Hout[b, m, :] = X[b, gidx[m], :] @ W
// One wave -> 16 x (NT*16) tile, K in steps of 4 via v_wmma_f32_16x16x4_f32.
// W (K x N, row-major) staged in LDS. 8 waves / 256-thread block.
// ============================================================
template <int NT>
__global__ __launch_bounds__(256)
void wmma_gemm_gather(const float* __restrict__ X, long xBatchStride,
                      const int* __restrict__ gidx,
                      const float* __restrict__ W,
                      float* __restrict__ Hout,
                      int M, int K)
{
  constexpr int N = NT * 16;
  __shared__ float sW[64 * 64];

  const int tid = threadIdx.x;
  for (int i = tid; i < K * N; i += blockDim.x) sW[i] = W[i];
  __syncthreads();

  const int wave = tid >> 5;
  const int lane = tid & 31;
  const int tilesPerBatch = (M + 15) >> 4;
  const int t = blockIdx.x * 8 + wave;   // uniform within wave -> EXEC stays all-ones
  const int b = blockIdx.y;
  if (t >= tilesPerBatch) return;

  const int half = lane >> 4;   // 0: lanes 0-15 (K,K+1), 1: lanes 16-31 (K+2,K+3)
  const int lid  = lane & 15;
  int row = t * 16 + lid;
  if (row >= M) row = M - 1;                    // clamp; stores are guarded below
  const int g = gidx ? gidx[row] : row;
  const float* arow = X + (long)b * xBatchStride + (long)g * K;

  v8f acc[NT];
#pragma unroll
  for (int i = 0; i < NT; ++i)
#pragma unroll
    for (int j = 0; j < 8; ++j) acc[i][j] = 0.0f;

  for (int k0 = 0; k0 < K; k0 += 4) {
    const int kk = k0 + half * 2;
    v2f a;
    a.x = arow[kk];
    a.y = arow[kk + 1];
#pragma unroll
    for (int n0 = 0; n0 < NT; ++n0) {
      const int col = n0 * 16 + lid;
      v2f bf;
      bf.x = sW[kk * N + col];
      bf.y = sW[(kk + 1) * N + col];
      acc[n0] = __builtin_amdgcn_wmma_f32_16x16x4_f32(
          false, a, false, bf, (short)0, acc[n0], false, false);
    }
  }

  // D layout: VGPR v holds M = v (lanes 0-15), M = v+8 (lanes 16-31); N = lid + n0*16
  float* obase = Hout + (long)b * M * N;
#pragma unroll
  for (int n0 = 0; n0 < NT; ++n0) {
    const int col = n0 * 16 + lid;
#pragma unroll
    for (int v = 0; v < 8; ++v) {
      const int r = t * 16 + half * 8 + v;
      if (r < M) obase[(long)r * N + col] = acc[n0][v];
    }
  }
}

// out[b, n, h] = dot(Hs[b, gidx(n), h*C : h*C+C], att[h, :])
__global__ void attn_coef_kernel(const float* __restrict__ Hs, int hsRows,
                                 const int* __restrict__ gidx,
                                 const float* __restrict__ att,
                                 float* __restrict__ outp,
                                 int Nrows, int H, int C)
{
  long i = (long)blockIdx.x * blockDim.x + threadIdx.x;
  long total = (long)B_ * Nrows * H;
  if (i >= total) return;
  int h = (int)(i % H);
  long t = i / H;
  int n = (int)(t % Nrows);
  int b = (int)(t / Nrows);
  int g = gidx ? gidx[n] : n;
  const float* hp = Hs + ((long)b * hsRows + g) * (long)(H * C) + h * C;
  const float* ap = att + h * C;
  float s = 0.f;
  for (int c = 0; c < C; ++c) s += hp[c] * ap[c];
  outp[i] = s;
}

__global__ void fill_u32_kernel(unsigned* __restrict__ p, unsigned v, long n)
{
  long i = (long)blockIdx.x * blockDim.x + threadIdx.x;
  if (i < n) p[i] = v;
}

// out[r*NC + c] = bias[c]
__global__ void fill_bias_kernel(float* __restrict__ p, const float* __restrict__ bias,
                                 int NC, long n)
{
  long i = (long)blockIdx.x * blockDim.x + threadIdx.x;
  if (i < n) p[i] = bias[i % NC];
}

__device__ __forceinline__ float edge_logit(const float* es, const float* ed,
                                            int b, int s, int d, int h,
                                            int H, int Nsrc, int Ndst)
{
  float v = es[((long)b * Nsrc + s) * H + h] + ed[((long)b * Ndst + d) * H + h];
  return v > 0.f ? v : NEG_SLOPE_ * v;
}

__global__ void edge_max_kernel(const float* __restrict__ es, const float* __restrict__ ed,
                                const int* __restrict__ src, const int* __restrict__ dst,
                                unsigned* __restrict__ menc,
                                int E, int H, int Nsrc, int Ndst)
{
  long i = (long)blockIdx.x * blockDim.x + threadIdx.x;
  long total = (long)B_ * E * H;
  if (i >= total) return;
  int h = (int)(i % H);
  long t = i / H;
  int e = (int)(t % E);
  int b = (int)(t / E);
  int s = src[e], d = dst[e];
  float v = edge_logit(es, ed, b, s, d, h, H, Nsrc, Ndst);
  atomicMax(&menc[((long)b * Ndst + d) * H + h], fenc(v));
}

__global__ void edge_sum_kernel(const float* __restrict__ es, const float* __restrict__ ed,
                                const int* __restrict__ src, const int* __restrict__ dst,
                                const unsigned* __restrict__ menc, float* __restrict__ den,
                                int E, int H, int Nsrc, int Ndst)
{
  long i = (long)blockIdx.x * blockDim.x + threadIdx.x;
  long total = (long)B_ * E * H;
  if (i >= total) return;
  int h = (int)(i % H);
  long t = i / H;
  int e = (int)(t % E);
  int b = (int)(t / E);
  int s = src[e], d = dst[e];
  float v = edge_logit(es, ed, b, s, d, h, H, Nsrc, Ndst);
  long didx = ((long)b * Ndst + d) * H + h;
  atomicAdd(&den[didx], expf(v - fdec(menc[didx])));
}

__global__ void edge_msg_kernel(const float* __restrict__ es, const float* __restrict__ ed,
                                const unsigned* __restrict__ menc, const float* __restrict__ den,
                                const float* __restrict__ Hs,
                                const int* __restrict__ src, const int* __restrict__ dst,
                                float* __restrict__ outp,
                                int E, int H, int C, int Nsrc, int Ndst)
{
  long i = (long)blockIdx.x * blockDim.x + threadIdx.x;
  long total = (long)B_ * E * H;
  if (i >= total) return;
  int h = (int)(i % H);
  long t = i / H;
  int e = (int)(t % E);
  int b = (int)(t / E);
  int s = src[e], d = dst[e];
  float v = edge_logit(es, ed, b, s, d, h, H, Nsrc, Ndst);
  long didx = ((long)b * Ndst + d) * H + h;
  float alpha = expf(v - fdec(menc[didx])) / den[didx];
  const float* hp = Hs + ((long)b * Nsrc + s) * (long)(H * C) + h * C;
  float* op = outp + ((long)b * Ndst + d) * (long)(H * C) + h * C;
  for (int c = 0; c < C; ++c) atomicAdd(&op[c], alpha * hp[c]);
}

__global__ void elu_kernel(float* __restrict__ p, long n)
{
  long i = (long)blockIdx.x * blockDim.x + threadIdx.x;
  if (i >= n) return;
  float x = p[i];
  p[i] = x > 0.f ? x : expm1f(x);
}

static inline int cdiv(long a, int b) { return (int)((a + (long)b - 1) / b); }

extern "C" void kernel_launch(void* const* d_in, const int* in_sizes, int n_in,
                              void* d_out, int out_size, void* d_ws, size_t ws_size,
                              hipStream_t stream)
{
  (void)in_sizes; (void)n_in; (void)out_size; (void)ws_size;

  const float* x     = (const float*)d_in[0];
  const int*   nid0  = (const int*)d_in[1];
  const int*   res0  = (const int*)d_in[2];
  const int*   esrc0 = (const int*)d_in[3];
  const int*   edst0 = (const int*)d_in[4];
  const int*   res1  = (const int*)d_in[5];
  const int*   esrc1 = (const int*)d_in[6];
  const int*   edst1 = (const int*)d_in[7];
  const float* W1    = (const float*)d_in[8];
  const float* as1   = (const float*)d_in[9];
  const float* ad1   = (const float*)d_in[10];
  const float* b1    = (const float*)d_in[11];
  const float* W2    = (const float*)d_in[12];
  const float* as2   = (const float*)d_in[13];
  const float* ad2   = (const float*)d_in[14];
  const float* b2    = (const float*)d_in[15];
  float* outp = (float*)d_out;

  // ---- workspace carve (all fp32/u32, ~64 MB) ----
  char* ws = (char*)d_ws;
  size_t off = 0;
  auto carve = [&](long nf) { void* p = ws + off; off += (size_t)nf * 4; return p; };
  float*    hs1  = (float*)   carve((long)B_ * N_SRC0_ * (H1_ * C1_)); // 7.68M
  float*    es1  = (float*)   carve((long)B_ * N_SRC0_ * H1_);
  float*    ed1  = (float*)   carve((long)B_ * N_DST0_ * H1_);
  unsigned* m1   = (unsigned*)carve((long)B_ * N_DST0_ * H1_);
  float*    den1 = (float*)   carve((long)B_ * N_DST0_ * H1_);
  float*    h1   = (float*)   carve((long)B_ * N_DST0_ * (H1_ * C1_)); // 3.84M
  float*    hs2  = (float*)   carve((long)B_ * N_DST0_ * (H2_ * C2_)); // 1.92M
  float*    es2  = (float*)   carve((long)B_ * N_DST0_ * H2_);
  float*    ed2  = (float*)   carve((long)B_ * N_DST1_ * H2_);
  unsigned* m2   = (unsigned*)carve((long)B_ * N_DST1_ * H2_);
  float*    den2 = (float*)   carve((long)B_ * N_DST1_ * H2_);

  const int TPB = 256;

  // ================= Layer 1 =================
  {
    const int tiles = (N_SRC0_ + 15) / 16;           // 1875
    dim3 grid(cdiv(tiles, 8), B_);
    wmma_gemm_gather<4><<<grid, 256, 0, stream>>>(
        x, (long)N_TOTAL_ * D_IN_, nid0, W1, hs1, N_SRC0_, D_IN_);
  }
  {
    long n = (long)B_ * N_SRC0_ * H1_;
    attn_coef_kernel<<<cdiv(n, TPB), TPB, 0, stream>>>(hs1, N_SRC0_, nullptr, as1, es1, N_SRC0_, H1_, C1_);
  }
  {
    long n = (long)B_ * N_DST0_ * H1_;
    attn_coef_kernel<<<cdiv(n, TPB), TPB, 0, stream>>>(hs1, N_SRC0_, res0, ad1, ed1, N_DST0_, H1_, C1_);
    fill_u32_kernel<<<cdiv(n, TPB), TPB, 0, stream>>>(m1, 0u, n);
    fill_u32_kernel<<<cdiv(n, TPB), TPB, 0, stream>>>((unsigned*)den1, 0u, n);
  }
  {
    long n = (long)B_ * N_DST0_ * (H1_ * C1_);
    fill_bias_kernel<<<cdiv(n, TPB), TPB, 0, stream>>>(h1, b1, H1_ * C1_, n);
  }
  {
    long n = (long)B_ * E0_ * H1_;
    edge_max_kernel<<<cdiv(n, TPB), TPB, 0, stream>>>(es1, ed1, esrc0, edst0, m1, E0_, H1_, N_SRC0_, N_DST0_);
    edge_sum_kernel<<<cdiv(n, TPB), TPB, 0, stream>>>(es1, ed1, esrc0, edst0, m1, den1, E0_, H1_, N_SRC0_, N_DST0_);
    edge_msg_kernel<<<cdiv(n, TPB), TPB, 0, stream>>>(es1, ed1, m1, den1, hs1, esrc0, edst0, h1,
                                                      E0_, H1_, C1_, N_SRC0_, N_DST0_);
  }
  {
    long n = (long)B_ * N_DST0_ * (H1_ * C1_);
    elu_kernel<<<cdiv(n, TPB), TPB, 0, stream>>>(h1, n);
  }

  // ================= Layer 2 =================
  {
    const int tiles = (N_DST0_ + 15) / 16;           // 938 (last tile partial, guarded)
    dim3 grid(cdiv(tiles, 8), B_);
    wmma_gemm_gather<2><<<grid, 256, 0, stream>>>(
        h1, (long)N_DST0_ * (H1_ * C1_), nullptr, W2, hs2, N_DST0_, H1_ * C1_);
  }
  {
    long n = (long)B_ * N_DST0_ * H2_;
    attn_coef_kernel<<<cdiv(n, TPB), TPB, 0, stream>>>(hs2, N_DST0_, nullptr, as2, es2, N_DST0_, H2_, C2_);
  }
  {
    long n = (long)B_ * N_DST1_ * H2_;
    attn_coef_kernel<<<cdiv(n, TPB), TPB, 0, stream>>>(hs2, N_DST0_, res1, ad2, ed2, N_DST1_, H2_, C2_);
    fill_u32_kernel<<<cdiv(n, TPB), TPB, 0, stream>>>(m2, 0u, n);
    fill_u32_kernel<<<cdiv(n, TPB), TPB, 0, stream>>>((unsigned*)den2, 0u, n);
  }
  {
    long n = (long)B_ * N_DST1_ * (H2_ * C2_);      // == out_size
    fill_bias_kernel<<<cdiv(n, TPB), TPB, 0, stream>>>(outp, b2, H2_ * C2_, n);
  }
  {
    long n = (long)B_ * E1_ * H2_;
    edge_max_kernel<<<cdiv(n, TPB), TPB, 0, stream>>>(es2, ed2, esrc1, edst1, m2, E1_, H2_, N_DST0_, N_DST1_);
    edge_sum_kernel<<<cdiv(n, TPB), TPB, 0, stream>>>(es2, ed2, esrc1, edst1, m2, den2, E1_, H2_, N_DST0_, N_DST1_);
    edge_msg_kernel<<<cdiv(n, TPB), TPB, 0, stream>>>(es2, ed2, m2, den2, hs2, esrc1, edst1, outp,
                                                      E1_, H2_, C2_, N_DST0_, N_DST1_);
  }
}